// SSIM3D_34213709480427
// MI455X (gfx1250) — compile-verified
//
#include <hip/hip_runtime.h>

// 3-D SSIM (window 11, sigma 1.5) on (4,1,128,128,128) fp32 volumes for gfx1250.
// Separable Gaussian blur as banded 16x16 GEMM tiles via V_WMMA_F32_16X16X4_F32.
// Problem is HBM-bound (~1 GB traffic @ 23.3 TB/s => ~40 us), so fp32 is free;
// WMMA is used to keep the conv arithmetic off the scalar/VALU critical path.
//
// This revision: compile-time field kind (no runtime ternary chain) and
// branchless SAME padding (clamped tap index + 0/1 mask) so inner loops are
// straight-line VALU/VMEM with EXEC all-ones — no s_cbranch in the hot path.
//
// ws layout (floats): [0, 8388608)        x-blur temp (reused per field)
//                     [(1+f)*8388608 ...) xy-blurred field f, f=0..4
// total = 6 * 8388608 * 4B = 192 MB.

typedef __attribute__((ext_vector_type(2))) float v2f;
typedef __attribute__((ext_vector_type(8))) float v8f;

#define VOL   2097152     // 128^3
#define PLANE 16384       // 128^2
#define NTOT  8388608     // 4 * 128^3

// Normalized Gaussian, sigma=1.5, taps d=-5..5 (index k = d+5).
__device__ __constant__ float c_GW[11] = {
    0.00102838f, 0.00759875f, 0.03600078f, 0.10936069f, 0.21300552f,
    0.26601172f,
    0.21300552f, 0.10936069f, 0.03600078f, 0.00759875f, 0.00102838f};

__device__ __forceinline__ float wgt(int k) {
  return ((unsigned)k < 11u) ? c_GW[k] : 0.0f;
}

__device__ __forceinline__ v8f conv_step(v8f acc, v2f a, v2f b) {
  // D = A(16x4) * B(4x16) + C  (8-arg: neg_a,A,neg_b,B,c_mod,C,reuse_a,reuse_b)
  return __builtin_amdgcn_wmma_f32_16x16x4_f32(false, a, false, b, (short)0, acc,
                                               false, false);
}

// Band-weight A matrix, 7 chunks of K=4: A[m, j] = w[j - m], m = lane&15.
// 32-bit A layout: lanes 0-15 hold K={4c,4c+1}, lanes 16-31 hold K={4c+2,4c+3}.
__device__ __forceinline__ void build_A(v2f A[7], int n, int hi) {
#pragma unroll
  for (int c = 0; c < 7; ++c) {
    const int j0 = 4 * c + 2 * hi;
    A[c][0] = wgt(j0 - n);
    A[c][1] = wgt(j0 + 1 - n);
  }
}

// Branchless SAME-pad helpers: clamp coord to [0,127], mask = in-range ? 1 : 0.
__device__ __forceinline__ int   clamp128(int q) { return min(max(q, 0), 127); }
__device__ __forceinline__ float mask128(int q)  {
  return ((unsigned)q < 128u) ? 1.0f : 0.0f;
}

// ---------------- Stage 1: conv along X of one on-the-fly field ----------------
// KIND: 0->i1, 1->i2, 2->i1*i1, 3->i2*i2, 4->i1*i2  (compile-time)
template <int KIND>
__global__ __launch_bounds__(256) void k_conv_x(const float* __restrict__ i1,
                                                const float* __restrict__ i2,
                                                float* __restrict__ out) {
  const int wave = threadIdx.x >> 5;
  const int lane = threadIdx.x & 31;
  const int t  = blockIdx.x * 8 + wave;      // 32768 tiles
  const int tx = t & 7;
  const int ty = (t >> 3) & 7;
  const int z  = (t >> 6) & 127;
  const int b  = t >> 13;
  const int n  = lane & 15;
  const int hi = lane >> 4;

  v2f A[7];
  build_A(A, n, hi);

  const int p0 = tx * 16;
  const int rowbase = b * VOL + z * PLANE + (ty * 16 + n) * 128;

  v8f acc = {};
#pragma unroll
  for (int c = 0; c < 7; ++c) {
    const int j0 = 4 * c + 2 * hi;
    v2f Bf;
#pragma unroll
    for (int s = 0; s < 2; ++s) {
      const int q  = p0 - 5 + j0 + s;        // input x, SAME zero pad
      const int qc = clamp128(q);
      float f;
      if (KIND == 0)      f = i1[rowbase + qc];
      else if (KIND == 1) f = i2[rowbase + qc];
      else if (KIND == 2) { const float a = i1[rowbase + qc]; f = a * a; }
      else if (KIND == 3) { const float a = i2[rowbase + qc]; f = a * a; }
      else { f = i1[rowbase + qc] * i2[rowbase + qc]; }
      Bf[s] = f * mask128(q);
    }
    acc = conv_step(acc, A[c], Bf);
  }

  // D: lane n = row y, VGPR r -> x offset r + 8*hi (contiguous -> float4 stores)
  const int xo = p0 + hi * 8;
  *reinterpret_cast<float4*>(out + rowbase + xo) =
      make_float4(acc[0], acc[1], acc[2], acc[3]);
  *reinterpret_cast<float4*>(out + rowbase + xo + 4) =
      make_float4(acc[4], acc[5], acc[6], acc[7]);
}

// ---------------- Stage 2: conv along Y ----------------
__global__ __launch_bounds__(256) void k_conv_y(const float* __restrict__ in,
                                                float* __restrict__ out) {
  const int wave = threadIdx.x >> 5;
  const int lane = threadIdx.x & 31;
  const int t  = blockIdx.x * 8 + wave;
  const int tx = t & 7;
  const int ty = (t >> 3) & 7;          // output y tile
  const int z  = (t >> 6) & 127;
  const int b  = t >> 13;
  const int n  = lane & 15;
  const int hi = lane >> 4;

  v2f A[7];
  build_A(A, n, hi);

  const int x  = tx * 16 + n;           // line = x (contiguous loads)
  const int p0 = ty * 16;
  const int bz = b * VOL + z * PLANE;

  v8f acc = {};
#pragma unroll
  for (int c = 0; c < 7; ++c) {
    const int j0 = 4 * c + 2 * hi;
    v2f Bf;
#pragma unroll
    for (int s = 0; s < 2; ++s) {
      const int q = p0 - 5 + j0 + s;    // input y
      Bf[s] = in[bz + clamp128(q) * 128 + x] * mask128(q);
    }
    acc = conv_step(acc, A[c], Bf);
  }

#pragma unroll
  for (int r = 0; r < 8; ++r)
    out[bz + (p0 + hi * 8 + r) * 128 + x] = acc[r];
}

// -------- Stage 3: conv along Z of all 5 fields + SSIM map + mean reduction --------
__global__ __launch_bounds__(256) void k_conv_z_ssim(
    const float* __restrict__ f1,  const float* __restrict__ f2,
    const float* __restrict__ f11, const float* __restrict__ f22,
    const float* __restrict__ f12, float* __restrict__ out) {
  const int wave = threadIdx.x >> 5;
  const int lane = threadIdx.x & 31;
  const int t  = blockIdx.x * 8 + wave;
  const int tx = t & 7;
  const int tz = (t >> 3) & 7;          // output z tile
  const int y  = (t >> 6) & 127;
  const int b  = t >> 13;
  const int n  = lane & 15;
  const int hi = lane >> 4;

  v2f A[7];
  build_A(A, n, hi);

  const int x  = tx * 16 + n;
  const int p0 = tz * 16;
  const int colbase = b * VOL + y * 128 + x;

  v8f a1 = {}, a2 = {}, a11 = {}, a22 = {}, a12 = {};
#pragma unroll
  for (int c = 0; c < 7; ++c) {
    const int j0 = 4 * c + 2 * hi;
    v2f B1, B2, B11, B22, B12;
#pragma unroll
    for (int s = 0; s < 2; ++s) {
      const int q   = p0 - 5 + j0 + s;  // input z
      const int idx = colbase + clamp128(q) * PLANE;
      const float m = mask128(q);
      B1[s]  = f1[idx]  * m;
      B2[s]  = f2[idx]  * m;
      B11[s] = f11[idx] * m;
      B22[s] = f22[idx] * m;
      B12[s] = f12[idx] * m;
    }
    a1  = conv_step(a1,  A[c], B1);
    a2  = conv_step(a2,  A[c], B2);
    a11 = conv_step(a11, A[c], B11);
    a22 = conv_step(a22, A[c], B22);
    a12 = conv_step(a12, A[c], B12);
  }

  const float C1v = 1e-4f;   // 0.01^2
  const float C2v = 9e-4f;   // 0.03^2
  float s = 0.0f;
#pragma unroll
  for (int r = 0; r < 8; ++r) {
    const float mu1 = a1[r], mu2 = a2[r];
    const float mu1s = mu1 * mu1, mu2s = mu2 * mu2, m12 = mu1 * mu2;
    const float s11 = a11[r] - mu1s;
    const float s22 = a22[r] - mu2s;
    const float s12 = a12[r] - m12;
    const float num = (2.0f * m12 + C1v) * (2.0f * s12 + C2v);
    const float den = (mu1s + mu2s + C1v) * (s11 + s22 + C2v);
    s += num / den;
  }

  // wave32 reduce -> LDS -> one atomic per block
#pragma unroll
  for (int off = 16; off; off >>= 1) s += __shfl_down(s, off, 32);
  __shared__ float red[8];
  if (lane == 0) red[wave] = s;
  __syncthreads();
  if (wave == 0) {
    float v = (lane < 8) ? red[lane] : 0.0f;
#pragma unroll
    for (int off = 4; off; off >>= 1) v += __shfl_down(v, off, 32);
    if (lane == 0) atomicAdd(out, v * (1.0f / (float)NTOT));
  }
}

__global__ void k_zero(float* o) { o[0] = 0.0f; }

extern "C" void kernel_launch(void* const* d_in, const int* in_sizes, int n_in,
                              void* d_out, int out_size, void* d_ws, size_t ws_size,
                              hipStream_t stream) {
  (void)in_sizes; (void)n_in; (void)out_size; (void)ws_size;
  const float* img1 = (const float*)d_in[0];
  const float* img2 = (const float*)d_in[1];
  // d_in[2] is window_size; fixed at 11 (device scalar unreadable under capture).
  float* out = (float*)d_out;
  float* ws  = (float*)d_ws;

  float* tmp = ws;                               // x-blur scratch (reused)
  float* fld[5];
  for (int f = 0; f < 5; ++f) fld[f] = ws + (size_t)(1 + f) * NTOT;

  k_zero<<<1, 1, 0, stream>>>(out);

  k_conv_x<0><<<4096, 256, 0, stream>>>(img1, img2, tmp);
  k_conv_y<<<4096, 256, 0, stream>>>(tmp, fld[0]);
  k_conv_x<1><<<4096, 256, 0, stream>>>(img1, img2, tmp);
  k_conv_y<<<4096, 256, 0, stream>>>(tmp, fld[1]);
  k_conv_x<2><<<4096, 256, 0, stream>>>(img1, img2, tmp);
  k_conv_y<<<4096, 256, 0, stream>>>(tmp, fld[2]);
  k_conv_x<3><<<4096, 256, 0, stream>>>(img1, img2, tmp);
  k_conv_y<<<4096, 256, 0, stream>>>(tmp, fld[3]);
  k_conv_x<4><<<4096, 256, 0, stream>>>(img1, img2, tmp);
  k_conv_y<<<4096, 256, 0, stream>>>(tmp, fld[4]);

  k_conv_z_ssim<<<4096, 256, 0, stream>>>(fld[0], fld[1], fld[2], fld[3], fld[4], out);
}